// CosineSimTierList_34325378629726
// MI455X (gfx1250) — compile-verified
//
#include <hip/hip_runtime.h>
#include <hip/hip_bf16.h>
#include <math.h>

typedef __attribute__((ext_vector_type(2))) float v2f;
typedef __attribute__((ext_vector_type(8))) float v8f;

// ---------------------------------------------------------------------------
// Kernel 1: normalize item vectors.  One wave (32 lanes) per batch row b.
// itemn[b,d] = item[b,d] / (||item[b,:]|| + 1e-8)
// ---------------------------------------------------------------------------
__global__ __launch_bounds__(32) void item_norm_kernel(
    const float* __restrict__ item, float* __restrict__ itemn)
{
    const int b = blockIdx.x;
    const int lane = threadIdx.x;          // 0..31
    const float v0 = item[b * 64 + lane];
    const float v1 = item[b * 64 + lane + 32];
    float ss = v0 * v0 + v1 * v1;
    #pragma unroll
    for (int o = 16; o > 0; o >>= 1) ss += __shfl_xor(ss, o);
    const float scale = 1.0f / (sqrtf(ss) + 1e-8f);
    itemn[b * 64 + lane]      = v0 * scale;
    itemn[b * 64 + lane + 32] = v1 * scale;
}

// ---------------------------------------------------------------------------
// Kernel 2: cosine via V_WMMA_F32_16X16X4_F32.
// One wave computes a 16-row tile: A = seq rows (16x4 per chunk, 16 chunks of
// K), B = normalized item replicated across all 16 N columns.  Row sum-of-
// squares computed in VALU on the same registers; cos = dot * 1/(||row||+eps),
// masked to -2.0 where indicator <= 0.
//
// A-fragment (16x4 f32): lane L<16 -> row L, K=0,1 ; lane L+16 -> row L, K=2,3
// B-fragment (4x16 f32): lanes 0-15 -> K=0 (x), K=1 (y); lanes 16-31 -> K=2,3
// D: vgpr j, half hi -> row m = 8*hi + j (all N identical here)
// ---------------------------------------------------------------------------
__global__ __launch_bounds__(256) void cos_wmma_kernel(
    const float* __restrict__ seq, const int* __restrict__ ind,
    const float* __restrict__ itemn, float* __restrict__ cosout,
    int S, int tilesPerB /* = S/128 */)
{
    __shared__ float lds_item[64];

    const int lane = threadIdx.x & 31;
    const int wave = threadIdx.x >> 5;     // 0..7
    const int blk  = blockIdx.x;
    const int b    = blk / tilesPerB;
    const int s0   = (blk % tilesPerB) * 128 + wave * 16;

    if (threadIdx.x < 64) lds_item[threadIdx.x] = itemn[b * 64 + threadIdx.x];
    __syncthreads();

    const int hi  = lane >> 4;             // which K-half this lane supplies
    const int row = lane & 15;             // M row handled by this lane
    const float* rowptr = seq + ((size_t)b * S + (s0 + row)) * 64;

    // Load 16 chunks (float2 each) of this row + partial sum of squares.
    v2f a[16];
    float ss = 0.0f;
    #pragma unroll
    for (int c = 0; c < 16; ++c) {
        const float2 t = *(const float2*)(rowptr + 4 * c + 2 * hi);
        a[c].x = t.x;
        a[c].y = t.y;
        ss = fmaf(t.x, t.x, ss);
        ss = fmaf(t.y, t.y, ss);
    }
    // Combine the two half-row partials: lanes L and L+16 own the same row.
    ss += __shfl_xor(ss, 16);
    const float inv = 1.0f / (sqrtf(ss) + 1e-8f);

    // WMMA accumulation chain over the 16 K-chunks.
    v8f acc = {0.f, 0.f, 0.f, 0.f, 0.f, 0.f, 0.f, 0.f};
    #pragma unroll
    for (int c = 0; c < 16; ++c) {
        v2f bfrag;
        bfrag.x = lds_item[4 * c + 2 * hi];
        bfrag.y = lds_item[4 * c + 2 * hi + 1];
        acc = __builtin_amdgcn_wmma_f32_16x16x4_f32(
            /*neg_a=*/false, a[c], /*neg_b=*/false, bfrag,
            /*c_mod=*/(short)0, acc, /*reuse_a=*/false, /*reuse_b=*/false);
    }

    // acc[j] on half hi = dot(seq row m, itemn) with m = 8*hi + j.
    // inv for row r lives on lanes r and r+16; shuffle BEFORE diverging
    // (disabled source lanes return 0 from ds_bpermute).
    float cosv[8];
    #pragma unroll
    for (int j = 0; j < 8; ++j) {
        const int m = 8 * hi + j;
        cosv[j] = acc[j] * __shfl(inv, m);
    }

    if (row == 0) {   // lanes 0 and 16 each emit 8 rows
        #pragma unroll
        for (int j = 0; j < 8; ++j) {
            const int m = 8 * hi + j;
            const size_t idx = (size_t)b * S + (s0 + m);
            const float c = (ind[idx] > 0) ? cosv[j] : -2.0f;
            cosout[idx] = c;
        }
    }
}

// ---------------------------------------------------------------------------
// Kernel 3: SimTier histogram + embedding expansion.  One block per batch b.
// ids = ceil(cos * (1/eps)) + bias; counts over valid bins; out =
// log(count+1) * emb, flattened [nb*8].
// ---------------------------------------------------------------------------
__global__ __launch_bounds__(256) void simtier_kernel(
    const float* __restrict__ cosv, const float* __restrict__ emb,
    float* __restrict__ out, int S, int nb, float inv_eps, int bias)
{
    __shared__ int counts[48];   // nb <= 42
    const int b = blockIdx.x;

    for (int i = threadIdx.x; i < nb; i += blockDim.x) counts[i] = 0;
    __syncthreads();

    for (int s = threadIdx.x; s < S; s += blockDim.x) {
        const float c = cosv[(size_t)b * S + s];
        const int id = (int)ceilf(c * inv_eps) + bias;
        if (id >= 0 && id < nb) atomicAdd(&counts[id], 1);
    }
    __syncthreads();

    const int n = nb * 8;
    for (int i = threadIdx.x; i < n; i += blockDim.x) {
        const float lc = logf((float)counts[i >> 3] + 1.0f);
        out[(size_t)b * n + i] = lc * emb[i];
    }
}

// ---------------------------------------------------------------------------
extern "C" void kernel_launch(void* const* d_in, const int* in_sizes, int n_in,
                              void* d_out, int out_size, void* d_ws, size_t ws_size,
                              hipStream_t stream)
{
    (void)in_sizes; (void)n_in; (void)out_size; (void)ws_size;

    const float* item      = (const float*)d_in[0];
    const float* seq_long  = (const float*)d_in[1];
    const float* seq_short = (const float*)d_in[2];
    const int*   ind_long  = (const int*)d_in[3];
    const int*   ind_short = (const int*)d_in[4];
    const float* emb_long  = (const float*)d_in[5];
    const float* emb_short = (const float*)d_in[6];

    const int B = 1024, SL = 2048, SS = 512;
    const int NBL = 22, NBS = 42, DIM = 8;

    float* out       = (float*)d_out;
    float* cos_long  = out;
    float* cos_short = cos_long  + (size_t)B * SL;
    float* st_long   = cos_short + (size_t)B * SS;
    float* st_short  = st_long   + (size_t)B * NBL * DIM;

    float* itemn = (float*)d_ws;   // [B, 64] normalized item vectors

    item_norm_kernel<<<B, 32, 0, stream>>>(item, itemn);

    cos_wmma_kernel<<<B * (SL / 128), 256, 0, stream>>>(
        seq_long, ind_long, itemn, cos_long, SL, SL / 128);
    cos_wmma_kernel<<<B * (SS / 128), 256, 0, stream>>>(
        seq_short, ind_short, itemn, cos_short, SS, SS / 128);

    simtier_kernel<<<B, 256, 0, stream>>>(cos_long,  emb_long,  st_long,  SL, NBL, 10.0f, 10);
    simtier_kernel<<<B, 256, 0, stream>>>(cos_short, emb_short, st_short, SS, NBS, 20.0f, 20);
}